// Aligner_50328426774586
// MI455X (gfx1250) — compile-verified
//
#include <hip/hip_runtime.h>
#include <hip/hip_bf16.h>

// Aligner pipeline for MI455X (gfx1250), f32-exact via V_WMMA_F32_16X16X4_F32.
// Round 3: permuted-K b128 operand loads (5 loads : 8 wmma), kw-major im2col
// with hoisted validity + pre-transposed (O,KW,CIN) weights — kills the
// per-element div/mod and halves VMEM instruction count per WMMA.

#define BATCH 16
#define TY    1600
#define TXQ   400
#define MELC  80
#define TXTC  512
#define MASC  384
#define HIDC  768
#define NEGV  (-1.0e9f)

typedef float v2f __attribute__((ext_vector_type(2)));
typedef float v4f __attribute__((ext_vector_type(4)));
typedef float v8f __attribute__((ext_vector_type(8)));

// D(16x16,f32) = A(16x4,f32) * B(4x16,f32) + C  — wave32, f32-exact multiplies.
__device__ __forceinline__ v8f wmma4(v2f a, v2f b, v8f c) {
  return __builtin_amdgcn_wmma_f32_16x16x4_f32(
      /*neg_a=*/false, a, /*neg_b=*/false, b,
      /*c_mod=*/(short)0, c, /*reuse_a=*/false, /*reuse_b=*/false);
}
__device__ __forceinline__ v2f lo2(v4f v) { v2f r; r.x = v.x; r.y = v.y; return r; }
__device__ __forceinline__ v2f hi2(v4f v) { v2f r; r.x = v.z; r.y = v.w; return r; }
__device__ __forceinline__ v4f ld4(const float* p) { return *(const v4f*)p; }

// Permuted-K 8-chunk: lanes 0-15 fetch logical K {k..k+3}, lanes 16-31 fetch
// {k+4..k+7}. Block0 consumes (lo2), block1 consumes (hi2) on BOTH A and B, so
// the reduction covers all 8 K-values exactly once — order-independent.

// ---------------------------------------------------------------------------
// Weight transpose prep: w (O, CIN, KW) -> wt (O, KW, CIN).
// ---------------------------------------------------------------------------
__global__ void aligner_wtrans(const float* __restrict__ w, float* __restrict__ wt,
                               int O, int CIN, int KW)
{
  const int n = O * CIN * KW;
  for (int i = blockIdx.x * blockDim.x + threadIdx.x; i < n;
       i += gridDim.x * blockDim.x) {
    const int o  = i / (CIN * KW);
    const int r  = i - o * (CIN * KW);
    const int ci = r / KW;
    const int kw = r - ci * KW;
    wt[(size_t)o * CIN * KW + (size_t)kw * CIN + ci] = w[i];
  }
}

// ---------------------------------------------------------------------------
// Kernel 1: conv1 (SAME, width KW) + bias + GLU (a * relu(g)) + keep mask.
// kw-major: validity hoisted per kw; inner ci loop is pure b128 + wmma.
// Each wave: 16 rows x 32 channels (2 tiles x (a,g) = 4 accs share one A load).
// ---------------------------------------------------------------------------
template<int CIN, int KW, int T>
__global__ void __launch_bounds__(192)
aligner_conv_glu(const float* __restrict__ x,    // (B,T,CIN)
                 const float* __restrict__ w1t,  // (2*HIDC, KW, CIN) transposed
                 const float* __restrict__ b1,   // (2*HIDC)
                 const int*   __restrict__ lens,
                 float* __restrict__ mid)        // (B,T,HIDC)
{
  constexpr int KRED = CIN * KW;
  constexpr int PAD  = KW / 2;
  const int lane = threadIdx.x & 31;
  const int wave = threadIdx.x >> 5;
  const int m0 = blockIdx.x * 16;                  // flat row tile over B*T
  const int n0 = blockIdx.y * 192 + wave * 32;     // 2 channel tiles: n0, n0+16
  const int b  = m0 / T;
  const int len = lens[b];
  const int l15 = lane & 15;
  const int kh  = (lane >> 4) * 4;                 // 0 (lanes 0-15) or 4 (16-31)
  const int tA = (m0 % T) + l15;
  const int co0 = n0 + l15;
  const int co1 = co0 + 16;
  const float* xb  = x + (size_t)b * T * CIN;
  const float* wA0 = w1t + (size_t)co0 * KRED;
  const float* wG0 = w1t + (size_t)(co0 + HIDC) * KRED;
  const float* wA1 = w1t + (size_t)co1 * KRED;
  const float* wG1 = w1t + (size_t)(co1 + HIDC) * KRED;

  v8f aA0 = {0.f,0.f,0.f,0.f,0.f,0.f,0.f,0.f};
  v8f aG0 = aA0, aA1 = aA0, aG1 = aA0;

#pragma unroll
  for (int kw = 0; kw < KW; ++kw) {
    const int tt = tA + kw - PAD;
    const bool okA = (tt >= 0) && (tt < T) && (tt < len);   // hoisted out of ci loop
    const float* pa  = xb + (size_t)(okA ? tt : 0) * CIN;
    const float* pA0 = wA0 + kw * CIN;
    const float* pG0 = wG0 + kw * CIN;
    const float* pA1 = wA1 + kw * CIN;
    const float* pG1 = wG1 + kw * CIN;
    for (int c0 = 0; c0 < CIN; c0 += 8) {
      v4f a4 = {0.f,0.f,0.f,0.f};
      if (okA) a4 = ld4(pa + c0 + kh);
      const v4f bA0 = ld4(pA0 + c0 + kh);
      const v4f bG0 = ld4(pG0 + c0 + kh);
      const v4f bA1 = ld4(pA1 + c0 + kh);
      const v4f bG1 = ld4(pG1 + c0 + kh);
      aA0 = wmma4(lo2(a4), lo2(bA0), aA0);
      aA0 = wmma4(hi2(a4), hi2(bA0), aA0);
      aG0 = wmma4(lo2(a4), lo2(bG0), aG0);
      aG0 = wmma4(hi2(a4), hi2(bG0), aG0);
      aA1 = wmma4(lo2(a4), lo2(bA1), aA1);
      aA1 = wmma4(hi2(a4), hi2(bA1), aA1);
      aG1 = wmma4(lo2(a4), lo2(bG1), aG1);
      aG1 = wmma4(hi2(a4), hi2(bG1), aG1);
    }
  }

  const int rowBase = m0 + (lane >> 4) * 8;
  const float bA0s = b1[co0], bG0s = b1[co0 + HIDC];
  const float bA1s = b1[co1], bG1s = b1[co1 + HIDC];
#pragma unroll
  for (int i = 0; i < 8; ++i) {
    const int row = rowBase + i;
    const int t = row % T;
    const float keep = (t < len) ? 1.f : 0.f;
    float a0 = aA0[i] + bA0s, g0 = aG0[i] + bG0s;
    float a1 = aA1[i] + bA1s, g1 = aG1[i] + bG1s;
    g0 = g0 > 0.f ? g0 : 0.f;
    g1 = g1 > 0.f ? g1 : 0.f;
    mid[(size_t)row * HIDC + co0] = a0 * g0 * keep;
    mid[(size_t)row * HIDC + co1] = a1 * g1 * keep;
  }
}

// ---------------------------------------------------------------------------
// Kernel 2: conv2 (SAME, width KW) + bias + keep mask.  kw-major, 4 N-tiles.
// One block (6 waves x 64 cols) covers N = 384.
// ---------------------------------------------------------------------------
template<int KW, int T>
__global__ void __launch_bounds__(192)
aligner_conv2(const float* __restrict__ mid, // (B,T,HIDC) (already keep-masked)
              const float* __restrict__ w2t, // (MASC, KW, HIDC) transposed
              const float* __restrict__ b2,  // (MASC)
              const int*   __restrict__ lens,
              float* __restrict__ proj)      // (B,T,MASC)
{
  constexpr int KRED = HIDC * KW;
  constexpr int PAD  = KW / 2;
  const int lane = threadIdx.x & 31;
  const int wave = threadIdx.x >> 5;
  const int m0 = blockIdx.x * 16;
  const int n0 = wave * 64;                        // 4 tiles: n0 .. n0+48
  const int b  = m0 / T;
  const int len = lens[b];
  const int l15 = lane & 15;
  const int kh  = (lane >> 4) * 4;
  const int tA = (m0 % T) + l15;
  const float* mb = mid + (size_t)b * T * HIDC;
  const float* wB0 = w2t + (size_t)(n0      + l15) * KRED;
  const float* wB1 = w2t + (size_t)(n0 + 16 + l15) * KRED;
  const float* wB2 = w2t + (size_t)(n0 + 32 + l15) * KRED;
  const float* wB3 = w2t + (size_t)(n0 + 48 + l15) * KRED;

  v8f a0 = {0.f,0.f,0.f,0.f,0.f,0.f,0.f,0.f};
  v8f a1 = a0, a2 = a0, a3 = a0;

#pragma unroll
  for (int kw = 0; kw < KW; ++kw) {
    const int tt = tA + kw - PAD;
    const bool okA = (tt >= 0) && (tt < T);          // mid already zero past len
    const float* pa = mb + (size_t)(okA ? tt : 0) * HIDC;
    const float* p0 = wB0 + kw * HIDC;
    const float* p1 = wB1 + kw * HIDC;
    const float* p2 = wB2 + kw * HIDC;
    const float* p3 = wB3 + kw * HIDC;
    for (int c0 = 0; c0 < HIDC; c0 += 8) {
      v4f a4 = {0.f,0.f,0.f,0.f};
      if (okA) a4 = ld4(pa + c0 + kh);
      const v4f b0 = ld4(p0 + c0 + kh);
      const v4f b1v = ld4(p1 + c0 + kh);
      const v4f b2v = ld4(p2 + c0 + kh);
      const v4f b3v = ld4(p3 + c0 + kh);
      a0 = wmma4(lo2(a4), lo2(b0),  a0);
      a0 = wmma4(hi2(a4), hi2(b0),  a0);
      a1 = wmma4(lo2(a4), lo2(b1v), a1);
      a1 = wmma4(hi2(a4), hi2(b1v), a1);
      a2 = wmma4(lo2(a4), lo2(b2v), a2);
      a2 = wmma4(hi2(a4), hi2(b2v), a2);
      a3 = wmma4(lo2(a4), lo2(b3v), a3);
      a3 = wmma4(hi2(a4), hi2(b3v), a3);
    }
  }

  const int rowBase = m0 + (lane >> 4) * 8;
  const float bb0 = b2[n0 + l15],      bb1 = b2[n0 + 16 + l15];
  const float bb2 = b2[n0 + 32 + l15], bb3 = b2[n0 + 48 + l15];
#pragma unroll
  for (int i = 0; i < 8; ++i) {
    const int row = rowBase + i;
    const int t = row % T;
    const float keep = (t < len) ? 1.f : 0.f;
    float* pr = proj + (size_t)row * MASC;
    pr[n0      + l15] = (a0[i] + bb0) * keep;
    pr[n0 + 16 + l15] = (a1[i] + bb1) * keep;
    pr[n0 + 32 + l15] = (a2[i] + bb2) * keep;
    pr[n0 + 48 + l15] = (a3[i] + bb3) * keep;
  }
}

// ---------------------------------------------------------------------------
// Kernel 3: out = in @ w^T + bias   (M rows, N = K = MASC).  b128 permuted-K.
// ---------------------------------------------------------------------------
__global__ void __launch_bounds__(192)
aligner_linear(const float* __restrict__ in,   // (M, MASC)
               const float* __restrict__ w,    // (MASC, MASC) row-major (out,in)
               const float* __restrict__ bias, // (MASC)
               float* __restrict__ out)        // (M, MASC)
{
  const int lane = threadIdx.x & 31;
  const int wave = threadIdx.x >> 5;
  const int m0 = blockIdx.x * 16;
  const int n0 = wave * 64;
  const int l15 = lane & 15;
  const int kh  = (lane >> 4) * 4;
  const float* inA = in + (size_t)(m0 + l15) * MASC;
  const float* wB0 = w + (size_t)(n0      + l15) * MASC;
  const float* wB1 = w + (size_t)(n0 + 16 + l15) * MASC;
  const float* wB2 = w + (size_t)(n0 + 32 + l15) * MASC;
  const float* wB3 = w + (size_t)(n0 + 48 + l15) * MASC;

  v8f a0 = {0.f,0.f,0.f,0.f,0.f,0.f,0.f,0.f};
  v8f a1 = a0, a2 = a0, a3 = a0;

  for (int k0 = 0; k0 < MASC; k0 += 8) {
    const v4f a4  = ld4(inA + k0 + kh);
    const v4f b0  = ld4(wB0 + k0 + kh);
    const v4f b1v = ld4(wB1 + k0 + kh);
    const v4f b2v = ld4(wB2 + k0 + kh);
    const v4f b3v = ld4(wB3 + k0 + kh);
    a0 = wmma4(lo2(a4), lo2(b0),  a0);
    a0 = wmma4(hi2(a4), hi2(b0),  a0);
    a1 = wmma4(lo2(a4), lo2(b1v), a1);
    a1 = wmma4(hi2(a4), hi2(b1v), a1);
    a2 = wmma4(lo2(a4), lo2(b2v), a2);
    a2 = wmma4(hi2(a4), hi2(b2v), a2);
    a3 = wmma4(lo2(a4), lo2(b3v), a3);
    a3 = wmma4(hi2(a4), hi2(b3v), a3);
  }

  const int rowBase = m0 + (lane >> 4) * 8;
  const float bb0 = bias[n0 + l15],      bb1 = bias[n0 + 16 + l15];
  const float bb2 = bias[n0 + 32 + l15], bb3 = bias[n0 + 48 + l15];
#pragma unroll
  for (int i = 0; i < 8; ++i) {
    float* pr = out + (size_t)(rowBase + i) * MASC;
    pr[n0      + l15] = a0[i] + bb0;
    pr[n0 + 16 + l15] = a1[i] + bb1;
    pr[n0 + 32 + l15] = a2[i] + bb2;
    pr[n0 + 48 + l15] = a3[i] + bb3;
  }
}

// ---------------------------------------------------------------------------
// Kernel 4: scores[b,y,x] = (q[b,y,:] . k[b,x,:]) / sqrt(MASC), both_pad mask.
// Each wave: 64 rows x 16 cols (4 M-tiles share one key b128 load).
// ---------------------------------------------------------------------------
__global__ void __launch_bounds__(32)
aligner_scores(const float* __restrict__ q,    // (B,TY,MASC)
               const float* __restrict__ kmat, // (B,TXQ,MASC)
               const int* __restrict__ xl, const int* __restrict__ yl,
               float* __restrict__ sc)         // (B,TY,TXQ)
{
  const int lane = threadIdx.x & 31;
  const int b  = blockIdx.z;
  const int m0 = blockIdx.x * 64;               // 4 y tiles
  const int n0 = blockIdx.y * 16;               // x tile
  const int l15 = lane & 15;
  const int kh  = (lane >> 4) * 4;
  const float* q0 = q + (size_t)(b * TY + m0      + l15) * MASC;
  const float* q1 = q + (size_t)(b * TY + m0 + 16 + l15) * MASC;
  const float* q2 = q + (size_t)(b * TY + m0 + 32 + l15) * MASC;
  const float* q3 = q + (size_t)(b * TY + m0 + 48 + l15) * MASC;
  const float* kb = kmat + (size_t)(b * TXQ + n0 + l15) * MASC;

  v8f a0 = {0.f,0.f,0.f,0.f,0.f,0.f,0.f,0.f};
  v8f a1 = a0, a2 = a0, a3 = a0;

  for (int k0 = 0; k0 < MASC; k0 += 8) {
    const v4f b4  = ld4(kb + k0 + kh);
    const v4f a40 = ld4(q0 + k0 + kh);
    const v4f a41 = ld4(q1 + k0 + kh);
    const v4f a42 = ld4(q2 + k0 + kh);
    const v4f a43 = ld4(q3 + k0 + kh);
    a0 = wmma4(lo2(a40), lo2(b4), a0);
    a0 = wmma4(hi2(a40), hi2(b4), a0);
    a1 = wmma4(lo2(a41), lo2(b4), a1);
    a1 = wmma4(hi2(a41), hi2(b4), a1);
    a2 = wmma4(lo2(a42), lo2(b4), a2);
    a2 = wmma4(hi2(a42), hi2(b4), a2);
    a3 = wmma4(lo2(a43), lo2(b4), a3);
    a3 = wmma4(hi2(a43), hi2(b4), a3);
  }

  const float scale = 0.05103103630798288f;  // 1/sqrt(384)
  const int tx = xl[b], ty = yl[b];
  const int xcol = n0 + l15;
  const bool xpad = (xcol >= tx);
  float* sb = sc + (size_t)b * TY * TXQ;
  const int rowBase = m0 + (lane >> 4) * 8;
#pragma unroll
  for (int i = 0; i < 8; ++i) {
    { const int y = rowBase + i;
      float v = a0[i] * scale; if (y >= ty && xpad) v = -1e-9f;
      sb[(size_t)y * TXQ + xcol] = v; }
    { const int y = rowBase + 16 + i;
      float v = a1[i] * scale; if (y >= ty && xpad) v = -1e-9f;
      sb[(size_t)y * TXQ + xcol] = v; }
    { const int y = rowBase + 32 + i;
      float v = a2[i] * scale; if (y >= ty && xpad) v = -1e-9f;
      sb[(size_t)y * TXQ + xcol] = v; }
    { const int y = rowBase + 48 + i;
      float v = a3[i] * scale; if (y >= ty && xpad) v = -1e-9f;
      sb[(size_t)y * TXQ + xcol] = v; }
  }
}

// ---------------------------------------------------------------------------
// Kernel 5: row softmax over x (400) in-place, plus logp = log(p + 1e-6).
// ---------------------------------------------------------------------------
__global__ void __launch_bounds__(128)
aligner_softmax_log(float* __restrict__ attn, float* __restrict__ lg)
{
  const int row = blockIdx.x;           // b*TY + y
  float* p  = attn + (size_t)row * TXQ;
  float* lp = lg   + (size_t)row * TXQ;
  const int tid = threadIdx.x;
  __shared__ float red[128];

  float v[4];
  float mx = -3.4e38f;
#pragma unroll
  for (int i = 0; i < 4; ++i) {
    const int idx = tid + i * 128;
    const float t = (idx < TXQ) ? p[idx] : -3.4e38f;
    v[i] = t;
    mx = fmaxf(mx, t);
  }
  red[tid] = mx;
  __syncthreads();
  for (int s = 64; s > 0; s >>= 1) {
    if (tid < s) red[tid] = fmaxf(red[tid], red[tid + s]);
    __syncthreads();
  }
  mx = red[0];
  __syncthreads();

  float sum = 0.f;
#pragma unroll
  for (int i = 0; i < 4; ++i) {
    const int idx = tid + i * 128;
    if (idx < TXQ) { v[i] = expf(v[i] - mx); sum += v[i]; }
  }
  red[tid] = sum;
  __syncthreads();
  for (int s = 64; s > 0; s >>= 1) {
    if (tid < s) red[tid] += red[tid + s];
    __syncthreads();
  }
  const float inv = 1.f / red[0];
#pragma unroll
  for (int i = 0; i < 4; ++i) {
    const int idx = tid + i * 128;
    if (idx < TXQ) {
      const float pv = v[i] * inv;
      p[idx]  = pv;
      lp[idx] = logf(pv + 1e-6f);     // safe_log
    }
  }
}

// ---------------------------------------------------------------------------
// Kernel 6: MAS (Glow-TTS maximum path). One block per batch item.
// ---------------------------------------------------------------------------
__global__ void __launch_bounds__(512)
aligner_mas(const float* __restrict__ lg,   // (B,TY,TXQ) attn_logprob
            const int* __restrict__ xl, const int* __restrict__ yl,
            float* __restrict__ hard,       // (B,TXQ,TY)
            float* __restrict__ dur,        // (B,TXQ)
            float* __restrict__ vbuf)       // (B,TY,TXQ) DP scratch
{
  const int b   = blockIdx.x;
  const int tid = threadIdx.x;
  const int tx = xl[b], ty = yl[b];
  float* hb = hard + (size_t)b * TXQ * TY;

  for (int i = tid; i < TXQ * TY; i += 512) hb[i] = 0.f;  // zero hard path

  __shared__ float sh[2][TXQ];
  __shared__ int durS[TXQ];
  if (tid < TXQ) {
    durS[tid] = 0;
    sh[0][tid] = (tid == 0) ? 0.f : NEGV;   // DP init column
  }
  __syncthreads();

  const float* lb = lg + (size_t)b * TY * TXQ;  // lp(x,y) = logp[b,y,x]
  float* vb = vbuf + (size_t)b * TY * TXQ;
  int pb = 0;
  for (int y = 0; y < TY; ++y) {
    if (tid < TXQ) {
      const bool valid = (tid < tx) && (y < ty) && (tid <= y);
      const float lpv = valid ? lb[(size_t)y * TXQ + tid] : NEGV;
      const float pl  = (tid > 0) ? sh[pb][tid - 1] : NEGV;
      const float cur = lpv + fmaxf(sh[pb][tid], pl);
      sh[pb ^ 1][tid] = cur;
      vb[(size_t)y * TXQ + tid] = cur;
    }
    __syncthreads();
    pb ^= 1;
  }

  if (tid == 0) {   // serial backtrack (reference bstep semantics)
    int xx = tx - 1;
    for (int y = TY - 1; y >= 0; --y) {
      const bool active = y < ty;
      if (active) {
        hb[(size_t)xx * TY + y] = 1.f;
        durS[xx] += 1;
        if (y > 0) {
          bool dec = false;
          if (xx > 0) {
            const float vpm = vb[(size_t)(y - 1) * TXQ + (xx - 1)];
            const float vpx = vb[(size_t)(y - 1) * TXQ + xx];
            dec = (xx == y) || (vpm > vpx);
          }
          if (dec) --xx;
        }
      }
    }
  }
  __syncthreads();
  if (tid < TXQ) dur[(size_t)b * TXQ + tid] = (float)durS[tid];
}

// ---------------------------------------------------------------------------
extern "C" void kernel_launch(void* const* d_in, const int* in_sizes, int n_in,
                              void* d_out, int out_size, void* d_ws, size_t ws_size,
                              hipStream_t stream)
{
  (void)in_sizes; (void)n_in; (void)out_size; (void)ws_size;
  const float* mel_x  = (const float*)d_in[0];
  const float* txt_x  = (const float*)d_in[1];
  const float* mel_w1 = (const float*)d_in[2];
  const float* mel_b1 = (const float*)d_in[3];
  const float* mel_w2 = (const float*)d_in[4];
  const float* mel_b2 = (const float*)d_in[5];
  const float* txt_w1 = (const float*)d_in[6];
  const float* txt_b1 = (const float*)d_in[7];
  const float* txt_w2 = (const float*)d_in[8];
  const float* txt_b2 = (const float*)d_in[9];
  const float* wq     = (const float*)d_in[10];
  const float* bq     = (const float*)d_in[11];
  const float* wk     = (const float*)d_in[12];
  const float* bk     = (const float*)d_in[13];
  const int* x_lens   = (const int*)d_in[14];
  const int* y_lens   = (const int*)d_in[15];

  // Output layout (flat, in return order): avg_attn, attn_logprob, attn_hard, dur
  float* avg  = (float*)d_out;
  float* logp = avg  + (size_t)BATCH * TY * TXQ;
  float* hard = logp + (size_t)BATCH * TY * TXQ;
  float* dur  = hard + (size_t)BATCH * TXQ * TY;

  // Workspace layout (floats): ~218 MB; MAS DP table reuses mel_mid.
  float* ws = (float*)d_ws;
  float* txt_mid  = ws;
  float* mel_mid  = txt_mid  + (size_t)BATCH * TXQ * HIDC;   // 4.92M f
  float* txt_proj = mel_mid  + (size_t)BATCH * TY  * HIDC;   // +19.66M f
  float* mel_proj = txt_proj + (size_t)BATCH * TXQ * MASC;   // +2.46M f
  float* qbuf     = mel_proj + (size_t)BATCH * TY  * MASC;   // +9.83M f
  float* kbuf     = qbuf     + (size_t)BATCH * TY  * MASC;   // +9.83M f
  float* w1t_txt  = kbuf     + (size_t)BATCH * TXQ * MASC;   // +2.46M f
  float* w1t_mel  = w1t_txt  + (size_t)(2 * HIDC) * TXTC * 3; // 2.36M f
  float* w2t_txt  = w1t_mel  + (size_t)(2 * HIDC) * MELC * 5; // 0.61M f
  float* w2t_mel  = w2t_txt  + (size_t)MASC * HIDC * 3;       // 0.88M f
  float* vbuf     = mel_mid;  // dead after conv2; holds (B,TY,TXQ) DP table

  // Weight transposes (O,CIN,KW) -> (O,KW,CIN)
  aligner_wtrans<<<512, 256, 0, stream>>>(txt_w1, w1t_txt, 2 * HIDC, TXTC, 3);
  aligner_wtrans<<<512, 256, 0, stream>>>(mel_w1, w1t_mel, 2 * HIDC, MELC, 5);
  aligner_wtrans<<<512, 256, 0, stream>>>(txt_w2, w2t_txt, MASC, HIDC, 3);
  aligner_wtrans<<<512, 256, 0, stream>>>(mel_w2, w2t_mel, MASC, HIDC, 5);

  aligner_conv_glu<TXTC,3,TXQ><<<dim3(BATCH*TXQ/16, HIDC/192), 192, 0, stream>>>(
      txt_x, w1t_txt, txt_b1, x_lens, txt_mid);
  aligner_conv_glu<MELC,5,TY><<<dim3(BATCH*TY/16, HIDC/192), 192, 0, stream>>>(
      mel_x, w1t_mel, mel_b1, y_lens, mel_mid);
  aligner_conv2<3,TXQ><<<dim3(BATCH*TXQ/16), 192, 0, stream>>>(
      txt_mid, w2t_txt, txt_b2, x_lens, txt_proj);
  aligner_conv2<5,TY><<<dim3(BATCH*TY/16), 192, 0, stream>>>(
      mel_mid, w2t_mel, mel_b2, y_lens, mel_proj);
  aligner_linear<<<dim3(BATCH*TY/16),  192, 0, stream>>>(mel_proj, wq, bq, qbuf);
  aligner_linear<<<dim3(BATCH*TXQ/16), 192, 0, stream>>>(txt_proj, wk, bk, kbuf);
  aligner_scores<<<dim3(TY/64, TXQ/16, BATCH), 32, 0, stream>>>(
      qbuf, kbuf, x_lens, y_lens, avg);
  aligner_softmax_log<<<dim3(BATCH*TY), 128, 0, stream>>>(avg, logp);
  aligner_mas<<<dim3(BATCH), 512, 0, stream>>>(logp, x_lens, y_lens, hard, dur, vbuf);
}